// MaxUnpooling2D_39668317946016
// MI455X (gfx1250) — compile-verified
//
#include <hip/hip_runtime.h>

// MaxUnpooling2D scatter-add for MI455X (gfx1250).
// pool: [16,128,128,64] f32 -> 2^24 elements, 2^20 per batch (2^18 float4s)
// ind : same shape, int32 (harness casts integer inputs to const int*)
// out : [16,256,256,64] f32 -> 2^22 elements (2^24 bytes) per batch
//
// Memory-bound scatter:
//   pass 1: zero 256MB output (b128 streaming stores)
//   pass 2: fire-and-forget global_atomic_add_f32 in GVS form
//           (saddr = uniform out base in SGPRs, per-lane signed 32-bit byte
//            offset in one VGPR; batch*2^24 + idx*4 < 2^31 so it fits).
// Atomics are no-return (STOREcnt only) and resolve at the L2 atomic units;
// the 16MB per-batch scatter window is L2-resident on the 192MB L2.
// s_endpgm's implicit wait-idle covers outstanding atomics.

static constexpr int TPB = 256;

__global__ __launch_bounds__(TPB) void zero_out_kernel(float4* __restrict__ out,
                                                       int nvec4) {
  int stride = (int)(gridDim.x * blockDim.x);
  for (int i = (int)(blockIdx.x * blockDim.x + threadIdx.x); i < nvec4;
       i += stride) {
    out[i] = make_float4(0.0f, 0.0f, 0.0f, 0.0f);  // global_store_b128
  }
}

__global__ __launch_bounds__(TPB) void scatter_add_kernel(
    const float4* __restrict__ pool, const int4* __restrict__ ind,
    float* __restrict__ out, int nvec4) {
  int stride = (int)(gridDim.x * blockDim.x);
  for (int i = (int)(blockIdx.x * blockDim.x + threadIdx.x); i < nvec4;
       i += stride) {
    // Keep next streaming tile in flight while this iteration's random
    // atomics drain (gfx1250 global_prefetch_b8; speculative, OOB-safe).
    __builtin_prefetch(pool + i + stride, 0, 0);
    __builtin_prefetch(ind + i + stride, 0, 0);

    float4 v = pool[i];  // global_load_b128, coalesced 512B/wave32
    int4 k = ind[i];     // global_load_b128, coalesced 512B/wave32

    // vec4 index -> batch: each batch owns 2^18 float4s; per-batch output
    // window is 2^24 bytes. Byte offset = batch<<24 | idx<<2 (< 2^31).
    unsigned bbase = ((unsigned)(i >> 18)) << 24;
    unsigned o0 = bbase + ((unsigned)k.x << 2);
    unsigned o1 = bbase + ((unsigned)k.y << 2);
    unsigned o2 = bbase + ((unsigned)k.z << 2);
    unsigned o3 = bbase + ((unsigned)k.w << 2);

    // GVS-form hardware FP32 atomic add, no-return:
    //   mem_addr = SGPR64(out) + VGPR_i32(offset)
    asm volatile("global_atomic_add_f32 %0, %1, %2" ::"v"(o0), "v"(v.x),
                 "s"(out)
                 : "memory");
    asm volatile("global_atomic_add_f32 %0, %1, %2" ::"v"(o1), "v"(v.y),
                 "s"(out)
                 : "memory");
    asm volatile("global_atomic_add_f32 %0, %1, %2" ::"v"(o2), "v"(v.z),
                 "s"(out)
                 : "memory");
    asm volatile("global_atomic_add_f32 %0, %1, %2" ::"v"(o3), "v"(v.w),
                 "s"(out)
                 : "memory");
  }
}

extern "C" void kernel_launch(void* const* d_in, const int* in_sizes, int n_in,
                              void* d_out, int out_size, void* d_ws,
                              size_t ws_size, hipStream_t stream) {
  const float* pool = (const float*)d_in[0];
  const int* ind = (const int*)d_in[1];
  float* out = (float*)d_out;

  // Pass 1: zero the 256MB output (required every call; harness poisons it).
  int nvec_out = out_size / 4;  // 16,777,216 float4 stores
  zero_out_kernel<<<4096, TPB, 0, stream>>>((float4*)out, nvec_out);

  // Pass 2: scatter-add.
  int nvec_in = in_sizes[0] / 4;  // 4,194,304 float4/int4 pairs
  scatter_add_kernel<<<4096, TPB, 0, stream>>>((const float4*)pool,
                                               (const int4*)ind, out, nvec_in);
}